// AtomLevelGNN_70342974374328
// MI455X (gfx1250) — compile-verified
//
#include <hip/hip_runtime.h>
#include <hip/hip_bf16.h>

#define NN    50000
#define EE    800000
#define EPN   (EE + NN)
#define DIN   64
#define HIDN  128
#define LAYERS 4
#define EPSV  1e-5f

typedef __attribute__((ext_vector_type(2))) float v2f;
typedef __attribute__((ext_vector_type(8))) float v8f;

// ---- monotonic float<->uint encoding for atomicMax over signed floats ----
__device__ __forceinline__ unsigned fenc(float f) {
    unsigned u = __float_as_uint(f);
    return (u & 0x80000000u) ? ~u : (u | 0x80000000u);
}
__device__ __forceinline__ float fdec(unsigned u) {
    u = (u & 0x80000000u) ? (u & 0x7FFFFFFFu) : ~u;
    return __uint_as_float(u);
}

// =====================================================================
// fp32 WMMA GEMM: out[nrows x 128] = A[nrows x K] * B[K x 128] (+bias,relu)
// grid.x = nrows/16, block = 128 (4 waves); wave w -> columns [32w,32w+32)
// Double-buffered LDS; W chunk stored K-pair interleaved so every B
// fragment is one aligned b64 LDS load; A tile staged with the CDNA5
// async-to-LDS path (GLOBAL_LOAD_ASYNC_TO_LDS_B128, ASYNCcnt).
// =====================================================================
__global__ __launch_bounds__(128)
void gemm_wmma_f32(const float* __restrict__ A, const float* __restrict__ B,
                   const float* __restrict__ bias, float* __restrict__ out,
                   int K, int relu_flag) {
    // Wl[bf]: 16 k-pairs x 128 cols x 2 -> Wl[bf][kp*256 + c*2 + (k&1)]
    __shared__ float Wl[2][16 * 256];   // 16 KB each
    __shared__ float Al[2][16 * 32];    // 2 KB each

    const int tid  = threadIdx.x;
    const int wave = tid >> 5;
    const int lane = tid & 31;
    const int half = lane >> 4;      // 0: lanes 0-15, 1: lanes 16-31
    const int m16  = lane & 15;
    const int row0 = blockIdx.x * 16;
    const int c0   = wave * 32;

    v8f acc0 = {};
    v8f acc1 = {};

    // stage one 32-deep K chunk (starting at kc) into buffer bf
    auto stage = [&](int kc, int bf) {
        // B chunk: 16 k-pairs x 128 cols; 512 col-groups-of-4, 4 per thread
        #pragma unroll
        for (int i = 0; i < 4; ++i) {
            int idx = tid + i * 128;         // 0..511
            int kp  = idx >> 5;              // 0..15
            int c4  = (idx & 31) << 2;       // 0..124
            const float* bp = B + (size_t)(kc + 2 * kp) * 128 + c4;
            float4 g0 = *(const float4*)bp;
            float4 g1 = *(const float4*)(bp + 128);
            float* w = &Wl[bf][kp * 256 + c4 * 2];
            *(float4*)(w + 0) = make_float4(g0.x, g1.x, g0.y, g1.y);
            *(float4*)(w + 4) = make_float4(g0.z, g1.z, g0.w, g1.w);
        }
        // A chunk: 16 rows x 32 cols = 128 lanes x 16B, async direct to LDS
        {
            int r = tid >> 3, cc = (tid & 7) << 2;
            unsigned lds = (unsigned)(uintptr_t)&Al[bf][r * 32 + cc];
            unsigned long long ga =
                (unsigned long long)(uintptr_t)&A[(size_t)(row0 + r) * K + kc + cc];
            asm volatile("global_load_async_to_lds_b128 %0, %1, off"
                         :: "v"(lds), "v"(ga) : "memory");
        }
    };

    stage(0, 0);
    asm volatile("s_wait_asynccnt 0x0" ::: "memory");
    __syncthreads();

    const int nch = K >> 5;
    for (int c = 0; c < nch; ++c) {
        const int bf = c & 1;
        if (c + 1 < nch) stage((c + 1) << 5, bf ^ 1);

        #pragma unroll
        for (int k0 = 0; k0 < 32; k0 += 4) {
            int kp = (k0 >> 1) + half;       // k-pair: rows kc+k0+2h, +1
            v2f a  = *(const v2f*)&Al[bf][m16 * 32 + k0 + 2 * half];
            v2f b0 = *(const v2f*)&Wl[bf][kp * 256 + (c0 + m16) * 2];
            v2f b1 = *(const v2f*)&Wl[bf][kp * 256 + (c0 + 16 + m16) * 2];
            acc0 = __builtin_amdgcn_wmma_f32_16x16x4_f32(
                       false, a, false, b0, (short)0, acc0, false, false);
            acc1 = __builtin_amdgcn_wmma_f32_16x16x4_f32(
                       false, a, false, b1, (short)0, acc1, false, false);
        }

        asm volatile("s_wait_asynccnt 0x0" ::: "memory");
        __syncthreads();
    }

    // C/D layout: VGPR j -> (row = j + 8*half, col = lane%16)
    #pragma unroll
    for (int j = 0; j < 8; ++j) {
        int row  = row0 + j + 8 * half;
        int col0 = c0 + m16;
        int col1 = c0 + 16 + m16;
        float v0 = acc0[j], v1 = acc1[j];
        if (bias) { v0 += bias[col0]; v1 += bias[col1]; }
        if (relu_flag) { v0 = fmaxf(v0, 0.f); v1 = fmaxf(v1, 0.f); }
        out[(size_t)row * 128 + col0] = v0;
        out[(size_t)row * 128 + col1] = v1;
    }
}

// ---------------- edge-attr mean: zero + reduce --------------------------
__global__ void zero12_kernel(float* easum) {
    if (threadIdx.x < 12) easum[threadIdx.x] = 0.f;
}

__global__ __launch_bounds__(256)
void easum_kernel(const float* __restrict__ eattr, float* __restrict__ easum) {
    __shared__ float sm[256 * 12];
    float acc[12];
    #pragma unroll
    for (int k = 0; k < 12; ++k) acc[k] = 0.f;
    for (int e = blockIdx.x * blockDim.x + threadIdx.x; e < EE;
         e += gridDim.x * blockDim.x) {
        #pragma unroll
        for (int k = 0; k < 12; ++k) acc[k] += eattr[(size_t)e * 12 + k];
    }
    #pragma unroll
    for (int k = 0; k < 12; ++k) sm[threadIdx.x * 12 + k] = acc[k];
    __syncthreads();
    for (int off = 128; off > 0; off >>= 1) {
        if (threadIdx.x < off) {
            #pragma unroll
            for (int k = 0; k < 12; ++k)
                sm[threadIdx.x * 12 + k] += sm[(threadIdx.x + off) * 12 + k];
        }
        __syncthreads();
    }
    if (threadIdx.x < 12) atomicAdd(&easum[threadIdx.x], sm[threadIdx.x]);
}

// wvec[l][hd][k] = sum_c We[l][k][hd*32+c]*att_edge[l][hd][c]; slat = mean_ea . wvec
__global__ void wvec_slat_kernel(const float* __restrict__ We,
                                 const float* __restrict__ att_edge,
                                 const float* __restrict__ easum,
                                 float* __restrict__ wvec,
                                 float* __restrict__ slat) {
    int i = threadIdx.x;
    if (i < LAYERS * 4 * 12) {
        int l = i / 48, rem = i % 48, hd = rem / 12, k = rem % 12;
        float t = 0.f;
        for (int c = 0; c < 32; ++c)
            t += We[(size_t)l * 12 * 128 + k * 128 + hd * 32 + c] *
                 att_edge[(size_t)l * 128 + hd * 32 + c];
        wvec[i] = t;
    }
    __syncthreads();
    if (i < LAYERS * 4) {
        int l = i / 4, hd = i % 4;
        float t = 0.f;
        for (int k = 0; k < 12; ++k)
            t += (easum[k] / (float)EE) * wvec[l * 48 + hd * 12 + k];
        slat[i] = t;
    }
}

// ---------------- per-layer init --------------------------
__global__ void layer_init_kernel(float* agg, unsigned* amaxu, float* denom,
                                  float* bnsum, float* bnsq) {
    size_t i = (size_t)blockIdx.x * blockDim.x + threadIdx.x;
    if (i < (size_t)NN * 128) agg[i] = 0.f;
    if (i < (size_t)NN * 4) { amaxu[i] = 0u; denom[i] = 0.f; }
    if (i < 128) { bnsum[i] = 0.f; bnsq[i] = 0.f; }
}

// a_src[n][hd], a_dst[n][hd] from xp
__global__ __launch_bounds__(256)
void node_att_kernel(const float* __restrict__ xp,
                     const float* __restrict__ a_s,
                     const float* __restrict__ a_d,
                     float* __restrict__ asrc, float* __restrict__ adst) {
    int n = blockIdx.x * blockDim.x + threadIdx.x;
    if (n >= NN) return;
    const float* xr = xp + (size_t)n * 128;
    #pragma unroll
    for (int hd = 0; hd < 4; ++hd) {
        float s = 0.f, d = 0.f;
        #pragma unroll
        for (int c = 0; c < 32; ++c) {
            float v = xr[hd * 32 + c];
            s += v * a_s[hd * 32 + c];
            d += v * a_d[hd * 32 + c];
        }
        asrc[n * 4 + hd] = s;
        adst[n * 4 + hd] = d;
    }
}

// edge logits + leaky relu + segment max (encoded-uint atomicMax)
__global__ __launch_bounds__(256)
void edge_logit_kernel(const int* __restrict__ ei, const float* __restrict__ eattr,
                       const float* __restrict__ wv, const float* __restrict__ sl,
                       const float* __restrict__ asrc, const float* __restrict__ adst,
                       unsigned* __restrict__ amaxu, float* __restrict__ elog) {
    int e = blockIdx.x * blockDim.x + threadIdx.x;
    if (e >= EPN) return;
    int s, d;
    float ef[4];
    if (e < EE) {
        s = ei[e]; d = ei[EE + e];
        float ea[12];
        #pragma unroll
        for (int k = 0; k < 12; ++k) ea[k] = eattr[(size_t)e * 12 + k];
        #pragma unroll
        for (int hd = 0; hd < 4; ++hd) {
            float t = 0.f;
            #pragma unroll
            for (int k = 0; k < 12; ++k) t += ea[k] * wv[hd * 12 + k];
            ef[hd] = t;
        }
    } else {
        s = e - EE; d = s;
        #pragma unroll
        for (int hd = 0; hd < 4; ++hd) ef[hd] = sl[hd];
    }
    #pragma unroll
    for (int hd = 0; hd < 4; ++hd) {
        float a = asrc[s * 4 + hd] + adst[d * 4 + hd] + ef[hd];
        a = (a > 0.f) ? a : 0.2f * a;          // leaky_relu(0.2)
        elog[(size_t)e * 4 + hd] = a;
        atomicMax(&amaxu[d * 4 + hd], fenc(a));
    }
}

// ex = exp(a - amax[dst]); segment-sum into denom
__global__ __launch_bounds__(256)
void edge_exp_kernel(const int* __restrict__ ei, const unsigned* __restrict__ amaxu,
                     float* __restrict__ elog, float* __restrict__ denom) {
    int e = blockIdx.x * blockDim.x + threadIdx.x;
    if (e >= EPN) return;
    int d = (e < EE) ? ei[EE + e] : (e - EE);
    #pragma unroll
    for (int hd = 0; hd < 4; ++hd) {
        float m  = fdec(amaxu[d * 4 + hd]);
        float ex = expf(elog[(size_t)e * 4 + hd] - m);
        elog[(size_t)e * 4 + hd] = ex;
        atomicAdd(&denom[d * 4 + hd], ex);
    }
}

// one 32-lane slice per edge; lane handles 4 consecutive channels
__global__ __launch_bounds__(256)
void scatter_kernel(const int* __restrict__ ei, const float* __restrict__ xp,
                    const float* __restrict__ elog, const float* __restrict__ denom,
                    float* __restrict__ agg) {
    size_t gid = (size_t)blockIdx.x * blockDim.x + threadIdx.x;
    int e    = (int)(gid >> 5);
    int lane = (int)(gid & 31);
    if (e >= EPN) return;
    int s = (e < EE) ? ei[e] : (e - EE);
    int d = (e < EE) ? ei[EE + e] : (e - EE);
    int hd = lane >> 3;                          // 8 lanes per head (32 ch)
    float alpha = elog[(size_t)e * 4 + hd] / denom[d * 4 + hd];
    float4 v = *(const float4*)&xp[(size_t)s * 128 + lane * 4];
    float* a = &agg[(size_t)d * 128 + lane * 4];
    atomicAdd(&a[0], alpha * v.x);
    atomicAdd(&a[1], alpha * v.y);
    atomicAdd(&a[2], alpha * v.z);
    atomicAdd(&a[3], alpha * v.w);
}

// per-channel sums / sums-of-squares over (agg + bias)
__global__ __launch_bounds__(256)
void bn_stats_kernel(const float* __restrict__ agg, const float* __restrict__ bias,
                     float* __restrict__ bnsum, float* __restrict__ bnsq) {
    __shared__ float s1[256], s2[256];
    int ch  = threadIdx.x & 127;
    int sub = threadIdx.x >> 7;
    float b = bias[ch];
    float sum = 0.f, sq = 0.f;
    for (int r = blockIdx.x * 2 + sub; r < NN; r += gridDim.x * 2) {
        float v = agg[(size_t)r * 128 + ch] + b;
        sum += v; sq += v * v;
    }
    s1[threadIdx.x] = sum; s2[threadIdx.x] = sq;
    __syncthreads();
    if (sub == 0) {
        atomicAdd(&bnsum[ch], s1[ch] + s1[128 + ch]);
        atomicAdd(&bnsq[ch],  s2[ch] + s2[128 + ch]);
    }
}

__global__ void bn_finalize_kernel(const float* bnsum, const float* bnsq,
                                   float* bnmu, float* bnistd) {
    int ch = threadIdx.x;
    if (ch >= 128) return;
    float mu  = bnsum[ch] / (float)NN;
    float var = bnsq[ch] / (float)NN - mu * mu;
    bnmu[ch]   = mu;
    bnistd[ch] = rsqrtf(var + EPSV);
}

// y = relu(gamma*(v-mu)*istd + beta) [+ h_in]
__global__ __launch_bounds__(256)
void bn_apply_kernel(const float* __restrict__ agg, const float* __restrict__ bias,
                     const float* __restrict__ gamma, const float* __restrict__ beta,
                     const float* __restrict__ bnmu, const float* __restrict__ bnistd,
                     const float* __restrict__ hin, float* __restrict__ out) {
    size_t i = (size_t)blockIdx.x * blockDim.x + threadIdx.x;
    if (i >= (size_t)NN * 128) return;
    int ch = (int)(i & 127);
    float v = agg[i] + bias[ch];
    float y = gamma[ch] * (v - bnmu[ch]) * bnistd[ch] + beta[ch];
    y = fmaxf(y, 0.f);
    if (hin) y += hin[i];
    out[i] = y;
}

// =====================================================================
extern "C" void kernel_launch(void* const* d_in, const int* in_sizes, int n_in,
                              void* d_out, int out_size, void* d_ws, size_t ws_size,
                              hipStream_t stream) {
    const float* x        = (const float*)d_in[0];
    const int*   ei       = (const int*)  d_in[1];   // [2,E]
    const float* eattr    = (const float*)d_in[2];   // [E,12]
    const float* Wp       = (const float*)d_in[3];   // [64,128]
    const float* bp       = (const float*)d_in[4];   // [128]
    const float* W        = (const float*)d_in[5];   // [L,128,128]
    const float* We       = (const float*)d_in[6];   // [L,12,128]
    const float* att_src  = (const float*)d_in[7];   // [L,4,32]
    const float* att_dst  = (const float*)d_in[8];
    const float* att_edge = (const float*)d_in[9];
    const float* bias     = (const float*)d_in[10];  // [L,128]
    const float* gamma    = (const float*)d_in[11];
    const float* beta     = (const float*)d_in[12];
    float* outp = (float*)d_out;

    // workspace layout (floats)
    float* ws = (float*)d_ws;
    const size_t NH = (size_t)NN * 128;
    float*    h0     = ws;                 // N*128
    float*    h1     = h0 + NH;            // N*128
    float*    xp     = h1 + NH;            // N*128
    float*    agg    = xp + NH;            // N*128
    float*    asrc   = agg + NH;           // N*4
    float*    adst   = asrc + (size_t)NN * 4;
    unsigned* amaxu  = (unsigned*)(adst + (size_t)NN * 4);   // N*4
    float*    denom  = (float*)(amaxu + (size_t)NN * 4);     // N*4
    float*    elog   = denom + (size_t)NN * 4;               // EP*4
    float*    wvec   = elog + (size_t)EPN * 4;               // L*48
    float*    slat   = wvec + LAYERS * 48;                   // L*4
    float*    easum  = slat + LAYERS * 4;                    // 16
    float*    bnsum  = easum + 16;                           // 128
    float*    bnsq   = bnsum + 128;
    float*    bnmu   = bnsq + 128;
    float*    bnistd = bnmu + 128;

    // ---- one-time precompute: edge-attr mean, wvec, self-loop att term ----
    zero12_kernel<<<1, 32, 0, stream>>>(easum);
    easum_kernel<<<256, 256, 0, stream>>>(eattr, easum);
    wvec_slat_kernel<<<1, 256, 0, stream>>>(We, att_edge, easum, wvec, slat);

    // ---- input projection: h0 = relu(x @ Wp + bp) ----
    gemm_wmma_f32<<<NN / 16, 128, 0, stream>>>(x, Wp, bp, h0, DIN, 1);

    const int TPB = 256;
    const int gN   = (NN + TPB - 1) / TPB;
    const int gNH  = (int)((NH + TPB - 1) / TPB);
    const int gEP  = (EPN + TPB - 1) / TPB;
    const int gSC  = (int)(((size_t)EPN * 32 + TPB - 1) / TPB);

    for (int l = 0; l < LAYERS; ++l) {
        float* hin  = (l & 1) ? h1 : h0;
        float* hout = (l == LAYERS - 1) ? outp : ((l & 1) ? h0 : h1);

        gemm_wmma_f32<<<NN / 16, 128, 0, stream>>>(
            hin, W + (size_t)l * 128 * 128, nullptr, xp, HIDN, 0);

        layer_init_kernel<<<gNH, TPB, 0, stream>>>(agg, amaxu, denom, bnsum, bnsq);

        node_att_kernel<<<gN, TPB, 0, stream>>>(
            xp, att_src + (size_t)l * 128, att_dst + (size_t)l * 128, asrc, adst);

        edge_logit_kernel<<<gEP, TPB, 0, stream>>>(
            ei, eattr, wvec + l * 48, slat + l * 4, asrc, adst, amaxu, elog);

        edge_exp_kernel<<<gEP, TPB, 0, stream>>>(ei, amaxu, elog, denom);

        scatter_kernel<<<gSC, TPB, 0, stream>>>(ei, xp, elog, denom, agg);

        bn_stats_kernel<<<256, TPB, 0, stream>>>(
            agg, bias + (size_t)l * 128, bnsum, bnsq);
        bn_finalize_kernel<<<1, 128, 0, stream>>>(bnsum, bnsq, bnmu, bnistd);

        bn_apply_kernel<<<gNH, TPB, 0, stream>>>(
            agg, bias + (size_t)l * 128, gamma + (size_t)l * 128,
            beta + (size_t)l * 128, bnmu, bnistd,
            (l > 0) ? hin : nullptr, hout);
    }
}